// CovarianceComplexBatchNorm_23862838297129
// MI455X (gfx1250) — compile-verified
//
#include <hip/hip_runtime.h>
#include <math.h>

typedef __attribute__((ext_vector_type(2))) float v2f;
typedef __attribute__((ext_vector_type(4))) float v4f;
typedef __attribute__((ext_vector_type(8))) float v8f;

#define NCHUNK 128
#define CBN_EPS 1e-5f

// ---------------------------------------------------------------------------
// Stage 1: per-chunk partial raw moments per feature. Each thread owns 4
// consecutive features -> global_load_b128 streaming (512 B/wave/instr).
// partials layout: [chunk][quantity(5)][F]
// ---------------------------------------------------------------------------
__global__ void cbn_stats_partial(const float* __restrict__ real,
                                  const float* __restrict__ imag,
                                  float* __restrict__ partials,
                                  int F, int rowsPerChunk) {
    const int f4    = (blockIdx.y * blockDim.x + threadIdx.x) * 4;
    const int chunk = blockIdx.x;
    const int row0  = chunk * rowsPerChunk;

    v4f sr = {}, si = {}, srr = {}, sii = {}, sri = {};
    const float* pr = real + (size_t)row0 * F + f4;
    const float* pi = imag + (size_t)row0 * F + f4;

#pragma unroll 4
    for (int r = 0; r < rowsPerChunk; ++r) {
        v4f x = *(const v4f*)(pr + (size_t)r * F);
        v4f y = *(const v4f*)(pi + (size_t)r * F);
        sr += x;
        si += y;
        srr += x * x;
        sii += y * y;
        sri += x * y;
    }

    size_t base = ((size_t)chunk * 5) * F + f4;
    *(v4f*)(partials + base + 0 * (size_t)F) = sr;
    *(v4f*)(partials + base + 1 * (size_t)F) = si;
    *(v4f*)(partials + base + 2 * (size_t)F) = srr;
    *(v4f*)(partials + base + 3 * (size_t)F) = sii;
    *(v4f*)(partials + base + 4 * (size_t)F) = sri;
}

// ---------------------------------------------------------------------------
// Stage 2: WMMA chunk reduction. One wave per (quantity, 16-feature group).
// A = all-ones 16x4  ->  every row of D = column sums of B (4 chunks x 16 feats).
// Chained accumulation over NCHUNK/4 v_wmma_f32_16x16x4_f32 ops.
// B layout per ISA (32-bit operands): lanes 0-15 hold K=0 (v0), K=1 (v1);
// lanes 16-31 hold K=2 (v0), K=3 (v1), N = lane & 15.
// D layout: VGPR0 lanes 0-15 = row M=0, N=lane.
// ---------------------------------------------------------------------------
__global__ void cbn_wmma_reduce(const float* __restrict__ partials,
                                float* __restrict__ sums, int F) {
    const int groups = F / 16;
    const int q      = blockIdx.x / groups;   // quantity 0..4
    const int g      = blockIdx.x % groups;   // feature group
    const int fbase  = g * 16;

    const int lane  = threadIdx.x;            // wave32, one wave per block
    const int n     = lane & 15;
    const int khalf = (lane >> 4) << 1;       // 0 for lanes 0-15, 2 for 16-31

    v2f a;
    a.x = 1.0f;
    a.y = 1.0f;
    v8f c = {};

    const size_t fidx = (size_t)fbase + n;
    for (int ch = 0; ch < NCHUNK; ch += 4) {
        v2f b;
        b.x = partials[((size_t)(ch + khalf + 0) * 5 + q) * F + fidx];
        b.y = partials[((size_t)(ch + khalf + 1) * 5 + q) * F + fidx];
        // D = A(ones) x B + C : accumulate column sums of the 4x16 tile.
        c = __builtin_amdgcn_wmma_f32_16x16x4_f32(
                /*neg_a=*/false, a, /*neg_b=*/false, b,
                /*c_mod=*/(short)0, c, /*reuse_a=*/false, /*reuse_b=*/false);
    }

    if (lane < 16) sums[(size_t)q * F + fbase + lane] = c[0];
}

// ---------------------------------------------------------------------------
// Stage 3: tiny per-feature kernel: raw moments -> whitening -> fused affine.
// coef layout: [P, Q, R, S, Cr, Ci] each of length F.
// out_r = P*r + Q*i + Cr ; out_i = R*r + S*i + Ci
// ---------------------------------------------------------------------------
__global__ void cbn_coef(const float* __restrict__ sums,
                         const float* __restrict__ g_rr,
                         const float* __restrict__ g_ri,
                         const float* __restrict__ g_ii,
                         const float* __restrict__ b_r,
                         const float* __restrict__ b_i,
                         float* __restrict__ coef, int N, int F) {
    const int f = blockIdx.x * blockDim.x + threadIdx.x;
    if (f >= F) return;

    const float invN = 1.0f / (float)N;
    const float mr = sums[0 * (size_t)F + f] * invN;
    const float mi = sums[1 * (size_t)F + f] * invN;
    const float Crr = fmaf(-mr, mr, sums[2 * (size_t)F + f] * invN) + CBN_EPS;
    const float Cii = fmaf(-mi, mi, sums[3 * (size_t)F + f] * invN) + CBN_EPS;
    const float Cri = fmaf(-mr, mi, sums[4 * (size_t)F + f] * invN);

    const float s   = sqrtf(fmaf(Crr, Cii, -Cri * Cri));
    const float t   = sqrtf(Crr + Cii + 2.0f * s);
    const float inv = 1.0f / (s * t);
    const float Wrr = (Cii + s) * inv;
    const float Wii = (Crr + s) * inv;
    const float Wri = -Cri * inv;

    const float grr = g_rr[f], gri = g_ri[f], gii = g_ii[f];
    const float P = fmaf(grr, Wrr, gri * Wri);
    const float Q = fmaf(grr, Wri, gri * Wii);
    const float R = fmaf(gri, Wrr, gii * Wri);
    const float S = fmaf(gri, Wri, gii * Wii);

    coef[0 * (size_t)F + f] = P;
    coef[1 * (size_t)F + f] = Q;
    coef[2 * (size_t)F + f] = R;
    coef[3 * (size_t)F + f] = S;
    coef[4 * (size_t)F + f] = b_r[f] - P * mr - Q * mi;
    coef[5 * (size_t)F + f] = b_i[f] - R * mr - S * mi;
}

// ---------------------------------------------------------------------------
// Stage 4: streaming apply pass. b128 loads; outputs stored with the
// non-temporal hint (never re-read -> don't pollute L2). Coefficients are
// hoisted into registers and reused across all rows handled by the block.
// ---------------------------------------------------------------------------
__global__ void cbn_apply(const float* __restrict__ real,
                          const float* __restrict__ imag,
                          const float* __restrict__ coef,
                          float* __restrict__ out_r,
                          float* __restrict__ out_i, int N, int F) {
    for (int f4 = threadIdx.x * 4; f4 < F; f4 += blockDim.x * 4) {
        const v4f P  = *(const v4f*)(coef + 0 * (size_t)F + f4);
        const v4f Q  = *(const v4f*)(coef + 1 * (size_t)F + f4);
        const v4f R  = *(const v4f*)(coef + 2 * (size_t)F + f4);
        const v4f S  = *(const v4f*)(coef + 3 * (size_t)F + f4);
        const v4f Cr = *(const v4f*)(coef + 4 * (size_t)F + f4);
        const v4f Ci = *(const v4f*)(coef + 5 * (size_t)F + f4);

        for (int row = blockIdx.x; row < N; row += gridDim.x) {
            const size_t off = (size_t)row * F + f4;
            const v4f r = *(const v4f*)(real + off);
            const v4f i = *(const v4f*)(imag + off);
            const v4f orr = P * r + Q * i + Cr;
            const v4f oii = R * r + S * i + Ci;
            __builtin_nontemporal_store(orr, (v4f*)(out_r + off));
            __builtin_nontemporal_store(oii, (v4f*)(out_i + off));
        }
    }
}

extern "C" void kernel_launch(void* const* d_in, const int* in_sizes, int n_in,
                              void* d_out, int out_size, void* d_ws, size_t ws_size,
                              hipStream_t stream) {
    const float* real = (const float*)d_in[0];
    const float* imag = (const float*)d_in[1];
    const float* g_rr = (const float*)d_in[2];
    const float* g_ri = (const float*)d_in[3];
    const float* g_ii = (const float*)d_in[4];
    const float* b_r  = (const float*)d_in[5];
    const float* b_i  = (const float*)d_in[6];

    const int F = in_sizes[2];            // 1024
    const int N = in_sizes[0] / F;        // 65536

    float* ws       = (float*)d_ws;
    float* partials = ws;                                   // NCHUNK*5*F floats
    float* sums     = partials + (size_t)NCHUNK * 5 * F;    // 5*F floats
    float* coef     = sums + (size_t)5 * F;                 // 6*F floats

    const int rowsPerChunk = N / NCHUNK;

    // Stage 1: partial moments (reads 2*N*F*4 bytes = 512 MB, b128 streaming).
    dim3 g1(NCHUNK, (F / 4 + 255) / 256);
    cbn_stats_partial<<<g1, 256, 0, stream>>>(real, imag, partials, F, rowsPerChunk);

    // Stage 2: WMMA chunk reduction (5 quantities x F/16 feature groups, 1 wave each).
    cbn_wmma_reduce<<<5 * (F / 16), 32, 0, stream>>>(partials, sums, F);

    // Stage 3: per-feature coefficient fold (tiny).
    cbn_coef<<<(F + 255) / 256, 256, 0, stream>>>(sums, g_rr, g_ri, g_ii, b_r, b_i,
                                                  coef, N, F);

    // Stage 4: streaming apply (reads 512 MB, writes 512 MB non-temporal).
    float* out_r = (float*)d_out;
    float* out_i = out_r + (size_t)N * F;
    int grid4 = N < 4096 ? N : 4096;
    cbn_apply<<<grid4, 256, 0, stream>>>(real, imag, coef, out_r, out_i, N, F);
}